// GivensRotationPerHead_27702539059905
// MI455X (gfx1250) — compile-verified
//
#include <hip/hip_runtime.h>
#include <math.h>

#define NUM_HEADS 32
#define DIM 128
#define NROT 16

typedef unsigned int v4u __attribute__((ext_vector_type(4)));
typedef int          v4i __attribute__((ext_vector_type(4)));
typedef int          v8i __attribute__((ext_vector_type(8)));
typedef float        v2f __attribute__((ext_vector_type(2)));
typedef float        v8f __attribute__((ext_vector_type(8)));

__global__ __launch_bounds__(256)
void givens_per_head_kernel(const float* __restrict__ thetas,
                            float* __restrict__ out) {
    __shared__ float ldsQ[DIM * DIM];            // 64 KB tile (of 320 KB/WGP)
    const int head = blockIdx.x;
    const int tid  = threadIdx.x;

    // ---- Phase 1: identity fill, branchless (no scratch spills)
    float4* lds4 = reinterpret_cast<float4*>(ldsQ);
    #pragma unroll
    for (int k = 0; k < 16; ++k) {
        int q   = tid + k * 256;                 // float4 index 0..4095
        int f   = q << 2;
        int row = f >> 7;
        int col = f & (DIM - 1);
        float4 v;
        v.x = (row == col    ) ? 1.0f : 0.0f;
        v.y = (row == col + 1) ? 1.0f : 0.0f;
        v.z = (row == col + 2) ? 1.0f : 0.0f;
        v.w = (row == col + 3) ? 1.0f : 0.0f;
        lds4[q] = v;
    }
    __syncthreads();

    // ---- Phase 2: wave 0 builds the active 17x17 block.
    // All pairs are (0, r+1) => final block is rank-1 below the diagonal:
    //   Q[j][col<j] = u[j]*w[col],  u[j] = -s_{j-1}*P_{j-1},  w[col] = s_{col-1}/P_col,
    //   w[0] = 1,  Q[0][col] = P16*w[col],  diag(j>=1) = c_{j-1}.
    // The 16x16 strictly-lower region is one V_WMMA_F32_16X16X4_F32 outer product.
    if (tid < 32) {
        const int lane = tid;
        const float* th = thetas + head * NROT;

        float P  = 1.0f;                         // prefix product of cosines
        float uA = 0.0f;                         // lane L<16: u[L+1]
        float wB = (lane == 0) ? 1.0f : 0.0f;    // lane L:    w[L]
        float cd = 0.0f;                         // lane L:    c_{L-1}
        #pragma unroll
        for (int r = 0; r < NROT; ++r) {
            float t = th[r];                     // wave-uniform scalar load
            float c = __cosf(t);
            float s = __sinf(t);
            if (lane == r)     uA = -s * P;      // u[r+1] = -s_r * P_r
            float Pn = P * c;                    // P_{r+1}
            if (lane == r + 1) { wB = s / Pn; cd = c; }
            P = Pn;                              // after loop: P = P16
        }

        // A: 16x4, only K=0 column = u[1..16] (lanes 0-15, VGPR0); rest zero.
        // B: 4x16, only K=0 row    = w[0..15] (lanes 0-15, VGPR0); rest zero.
        v2f A; A.x = (lane < 16) ? uA : 0.0f; A.y = 0.0f;
        v2f B; B.x = (lane < 16) ? wB : 0.0f; B.y = 0.0f;
        v8f C = (v8f)0.0f;
        v8f D = __builtin_amdgcn_wmma_f32_16x16x4_f32(
            /*neg_a=*/false, A, /*neg_b=*/false, B,
            /*c_mod=*/(short)0, C, /*reuse_a=*/false, /*reuse_b=*/false);

        // D layout: VGPR g, lane L -> M = g + 8*(L>=16), N = L&15; row j = M+1.
        const int n     = lane & 15;
        const int mbase = (lane >> 4) * 8;
        #pragma unroll
        for (int g = 0; g < 8; ++g) {
            int m = g + mbase;
            float val = (n <= m) ? D[g] : 0.0f;  // keep strictly-lower (col < j)
            ldsQ[(m + 1) * DIM + n] = val;
        }
        // Diagonal c_{j-1} at (j,j), j=1..16 (same-wave LDS ops stay in order,
        // so this overwrites the zeros the masked store put on the diagonal).
        if (lane >= 1 && lane <= 16) ldsQ[lane * DIM + lane] = cd;
        // Row 0 = P16 * w  (lanes 17..31 have w=0 -> rewrite zeros, harmless).
        ldsQ[lane] = P * wB;
    }
    __syncthreads();

    // ---- Phase 3a: drain tile to HBM, coalesced 128-bit stores (correct path)
    float4* out4 = reinterpret_cast<float4*>(out) + (size_t)head * (DIM * DIM / 4);
    #pragma unroll
    for (int k = 0; k < 16; ++k) {
        int q = tid + k * 256;
        out4[q] = lds4[q];
    }

    // ---- Phase 3b: wave 0 issues one TDM store of the whole 64 KB tile.
    // Byte-identical data to phase 3a, so the duplicate write is order-safe.
    if (tid < 32) {
        unsigned long long gaddr =
            (unsigned long long)(out + (size_t)head * DIM * DIM);
        unsigned int ldsaddr =
            (unsigned int)(unsigned long long)(uintptr_t)(&ldsQ[0]);

        v4u g0;
        g0.x = 1u;                                   // count=1 (valid), user mode
        g0.y = ldsaddr;                              // lds_addr
        g0.z = (unsigned int)(gaddr & 0xFFFFFFFFu);  // global_addr[31:0]
        g0.w = (unsigned int)((gaddr >> 32) & 0x1FFFFFFu)
             | 0x80000000u;                          // type=2, bits 127:126

        v8i g1;
        g1[0] = (int)(2u << 16);       // mask=0; data_size=2 -> 4 bytes
        g1[1] = (int)(128u << 16);     // tensor_dim0[15:0] = 128
        g1[2] = (int)(128u << 16);     // tensor_dim1[15:0] = 128
        g1[3] = (int)(128u << 16);     // tile_dim0 = 128
        g1[4] = (int)(128u);           // tile_dim1 = 128; tile_dim2 = 0
        g1[5] = (int)(128u);           // tensor_dim0_stride = 128 elements
        g1[6] = 0;
        g1[7] = 0;

        v4i g2 = (v4i)0;
        v4i g3 = (v4i)0;

#if defined(__clang_major__) && (__clang_major__ >= 23)
        v8i g1b = (v8i)0;
        __builtin_amdgcn_tensor_store_from_lds(g0, g1, g2, g3, g1b, 0);
#else
        __builtin_amdgcn_tensor_store_from_lds(g0, g1, g2, g3, 0);
#endif
        __builtin_amdgcn_s_wait_tensorcnt(0);
    }
}

extern "C" void kernel_launch(void* const* d_in, const int* in_sizes, int n_in,
                              void* d_out, int out_size, void* d_ws, size_t ws_size,
                              hipStream_t stream) {
    const float* thetas = (const float*)d_in[0];   // (32, 16) f32
    float* out = (float*)d_out;                    // (32, 128, 128) f32
    (void)in_sizes; (void)n_in; (void)out_size; (void)d_ws; (void)ws_size;
    givens_per_head_kernel<<<NUM_HEADS, 256, 0, stream>>>(thetas, out);
}